// Advection2D_25082609009397
// MI455X (gfx1250) — compile-verified
//
#include <hip/hip_runtime.h>
#include <cstdint>

// ---------------- Tile geometry ----------------
#define TX 32                    // output tile width  (one wave per row)
#define TY 8                     // output tile height (8 waves / 256 thr)
#define HALO 3                   // WENO5 needs i-3..i+3 around each cell
#define SH_W   (TX + 2 * HALO)   // 38 staged columns
#define SH_R   (TY + 2 * HALO)   // 14 staged rows
#define SH_STRIDE (SH_W + 1)     // 39, odd stride to dodge bank conflicts

__device__ __forceinline__ float fast_rcp(float x) {
    return __builtin_amdgcn_rcpf(x);   // v_rcp_f32 (TRANS), ~1 ulp
}

// Classic WENO5 left-biased reconstruction at face i+1/2.
__device__ __forceinline__ float weno5(float qm2, float qm1, float q0,
                                       float qp1, float qp2) {
    const float EPS = 1e-8f;
    float f1 = (1.0f / 3.0f) * qm2 - (7.0f / 6.0f) * qm1 + (11.0f / 6.0f) * q0;
    float f2 = -(1.0f / 6.0f) * qm1 + (5.0f / 6.0f) * q0 + (1.0f / 3.0f) * qp1;
    float f3 = (1.0f / 3.0f) * q0 + (5.0f / 6.0f) * qp1 - (1.0f / 6.0f) * qp2;
    const float k1 = 13.0f / 12.0f, k2 = 0.25f;
    float d1 = qm2 - 2.0f * qm1 + q0;
    float e1 = qm2 - 4.0f * qm1 + 3.0f * q0;
    float b1 = k1 * d1 * d1 + k2 * e1 * e1;
    float d2 = qm1 - 2.0f * q0 + qp1;
    float e2 = qm1 - qp1;
    float b2 = k1 * d2 * d2 + k2 * e2 * e2;
    float d3 = q0 - 2.0f * qp1 + qp2;
    float e3 = 3.0f * q0 - 4.0f * qp1 + qp2;
    float b3 = k1 * d3 * d3 + k2 * e3 * e3;
    float t1 = b1 + EPS, t2 = b2 + EPS, t3 = b3 + EPS;
    // weights are normalized below, so fast rcp is accuracy-safe
    float w1 = 0.1f * fast_rcp(t1 * t1);
    float w2 = 0.6f * fast_rcp(t2 * t2);
    float w3 = 0.3f * fast_rcp(t3 * t3);
    return (w1 * f1 + w2 * f2 + w3 * f3) * fast_rcp(w1 + w2 + w3);
}

// gfx1250 async global->LDS copy (32-bit per active lane).
// VDST operand carries the LDS byte address, VADDR the 64-bit global address.
__device__ __forceinline__ void async_ld_f32(uint32_t lds_off, const float* g) {
    asm volatile("global_load_async_to_lds_b32 %0, %1, off"
                 :: "v"(lds_off), "v"(g)
                 : "memory");
}

__global__ __launch_bounds__(256) void adv2d_weno5_kernel(
    const float* __restrict__ h, const float* __restrict__ u,
    const float* __restrict__ v, float* __restrict__ out,
    int Ny, int Nx) {
    __shared__ float sh_h[SH_R][SH_STRIDE];   // staged h tile + halo
    __shared__ float sh_fe[TY][TX + 2];       // east-face fluxes, f in [0,33)
    __shared__ float sh_fn[TY + 1][TX];       // north-face fluxes, g in [0,9)

    const int tid = threadIdx.x;
    const int i0 = blockIdx.x * TX;
    const int j0 = blockIdx.y * TY;

    // ---- Stage h tile (edge-clamped halo) via async global->LDS ----
    const int n_stage = SH_R * SH_W;  // 532 elements, 256 threads
    for (int idx = tid; idx < n_stage; idx += 256) {
        int r = idx / SH_W;
        int c = idx - r * SH_W;
        int gj = j0 - HALO + r; gj = gj < 0 ? 0 : (gj >= Ny ? Ny - 1 : gj);
        int gi = i0 - HALO + c; gi = gi < 0 ? 0 : (gi >= Nx ? Nx - 1 : gi);
        uint32_t lds_off = (uint32_t)(uintptr_t)(&sh_h[r][c]);
        async_ld_f32(lds_off, h + (size_t)gj * Nx + gi);
    }
    asm volatile("s_wait_asynccnt 0" ::: "memory");
    __syncthreads();

    // ---- East-face fluxes: fe(j0+r, i0-1+f), r in [0,8), f in [0,33) ----
    // LDS column of global cell (i0-1+f)+k is (f+2)+k; stencil k=-2..3 -> cols f..f+5.
    for (int t = tid; t < TY * (TX + 1); t += 256) {   // 264 faces
        int r = t / (TX + 1);
        int f = t - r * (TX + 1);
        int jf = j0 + r;
        int iF = i0 + f - 1;
        int iC = iF < 0 ? 0 : (iF >= Nx ? Nx - 1 : iF);  // only OOB for don't-care faces
        float vel = u[(size_t)jf * Nx + iC];
        const float* row = &sh_h[r + HALO][f];
        float a0 = row[0], a1 = row[1], a2 = row[2];
        float a3 = row[3], a4 = row[4], a5 = row[5];
        bool pos = vel >= 0.0f;
        // pos: (h[i-2..i+2]); neg (mirrored): (h[i+3],h[i+2],h[i+1],h[i],h[i-1])
        float qm2 = pos ? a0 : a5;
        float qm1 = pos ? a1 : a4;
        float q0  = pos ? a2 : a3;
        float qp1 = pos ? a3 : a2;
        float qp2 = pos ? a4 : a1;
        sh_fe[r][f] = vel * weno5(qm2, qm1, q0, qp1, qp2);
    }

    // ---- North-face fluxes: fn(j0-1+g, i0+c), g in [0,9), c in [0,32) ----
    for (int t = tid; t < (TY + 1) * TX; t += 256) {   // 288 faces
        int g = t >> 5;            // t / 32
        int c = t & (TX - 1);
        int jF = j0 + g - 1;
        int jC = jF < 0 ? 0 : (jF >= Ny ? Ny - 1 : jF);
        float vel = v[(size_t)jC * Nx + (i0 + c)];
        int lc = c + HALO;
        float a0 = sh_h[g + 0][lc], a1 = sh_h[g + 1][lc], a2 = sh_h[g + 2][lc];
        float a3 = sh_h[g + 3][lc], a4 = sh_h[g + 4][lc], a5 = sh_h[g + 5][lc];
        bool pos = vel >= 0.0f;
        float qm2 = pos ? a0 : a5;
        float qm1 = pos ? a1 : a4;
        float q0  = pos ? a2 : a3;
        float qp1 = pos ? a3 : a2;
        float qp2 = pos ? a4 : a1;
        sh_fn[g][c] = vel * weno5(qm2, qm1, q0, qp1, qp2);
    }
    __syncthreads();

    // ---- Divergence on interior; zero on the 2-cell ghost ring ----
    const float RDX = 1.0f / 1000.0f;   // 1/DX
    const float RDY = 1.0f / 1000.0f;   // 1/DY
    int lx = tid & (TX - 1);
    int ly = tid >> 5;
    int gi = i0 + lx;
    int gj = j0 + ly;
    float res = 0.0f;
    if (gi >= 2 && gi <= Nx - 3 && gj >= 2 && gj <= Ny - 3) {
        float fe_e = sh_fe[ly][lx + 1];       // fe(j, i)
        float fe_w = sh_fe[ly][lx];           // fe(j, i-1)
        float fn_n = sh_fn[ly + 1][lx];       // fn(j, i)
        float fn_s = sh_fn[ly][lx];           // fn(j-1, i)
        res = -((fe_e - fe_w) * RDX + (fn_n - fn_s) * RDY);
    }
    out[(size_t)gj * Nx + gi] = res;
}

extern "C" void kernel_launch(void* const* d_in, const int* in_sizes, int n_in,
                              void* d_out, int out_size, void* d_ws, size_t ws_size,
                              hipStream_t stream) {
    const float* h = (const float*)d_in[0];
    const float* u = (const float*)d_in[1];
    const float* v = (const float*)d_in[2];
    float* out = (float*)d_out;
    const int Ny = 2048;
    const int Nx = 4096;
    dim3 grid(Nx / TX, Ny / TY);   // 128 x 256 blocks, exact tiling
    adv2d_weno5_kernel<<<grid, 256, 0, stream>>>(h, u, v, out, Ny, Nx);
}